// MAWAttention_12515534701251
// MI455X (gfx1250) — compile-verified
//
#include <hip/hip_runtime.h>
#include <math.h>

// ---------------------------------------------------------------------------
// MAW multi-depth attention for gfx1250 (MI455X), fp32 WMMA 16x16x4 path.
// Two-pass flash-style: pass1 computes softmax state (row max m, row sum Z)
// plus per-(b,d) gating statistics; pass2 builds depth gate weights; pass3
// recomputes p = exp(s-m)/Z tiles and produces context (p·V via WMMA) and
// the depth-mixed attention map. The 1/sqrt(ds) score scale is folded into
// the Q (A) fragments so WMMA outputs are pre-scaled (no per-tile VALU mul).
// ---------------------------------------------------------------------------

typedef __attribute__((ext_vector_type(2))) float v2f;
typedef __attribute__((ext_vector_type(8))) float v8f;

#define WMMA4(a, b, c) \
  __builtin_amdgcn_wmma_f32_16x16x4_f32(false, (a), false, (b), (short)0, (c), false, false)

static constexpr int B  = 4;
static constexpr int H  = 16;
static constexpr int L  = 512;
static constexpr int Dh = 64;
static constexpr int DEP = 8;
static constexpr int DS  = 8;                 // Dh / DEP
static constexpr float SCALE = 0.35355339059327373f;  // 1/sqrt(8)
static constexpr int QT = L / 16;             // 32 q-tiles
static constexpr int NBLK1 = B * H * DEP * QT; // 16384 pass1 blocks

// A/B fragment layout for V_WMMA_F32_16X16X4_F32 (per ISA 7.12.2):
//   lane l (row/col = l&15, hi = l>>4): v0 holds K = 2*hi, v1 holds K = 2*hi+1.

// ---------------------------------------------------------------------------
// Pass 1: per (b,h,d,q-tile): scaled-score strip (16x512) in LDS via WMMA,
// softmax state and row statistics. Deterministic partial sums (no atomics).
// ---------------------------------------------------------------------------
__global__ void __launch_bounds__(32)
maw_pass1(const float* __restrict__ Q, const float* __restrict__ Kc,
          float* __restrict__ rowM, float* __restrict__ rowZ,
          float* __restrict__ part)
{
  __shared__ float S[16 * L];   // 32 KB scaled-score strip

  const unsigned bid = blockIdx.x;
  const int qt = bid & 31;
  const int d  = (bid >> 5) & 7;
  const int h  = (bid >> 8) & 15;
  const int b  = bid >> 12;
  const int q0 = qt * 16;
  const int c0 = d * DS;

  const int lane = threadIdx.x;
  const int row  = lane & 15;
  const int hi   = lane >> 4;

  const float* qbase = Q  + ((size_t)(b * H + h) * L + q0) * Dh;
  const float* kmat  = Kc + ((size_t)(b * H + h) * L) * Dh;

  // A fragments (Q tile, K-chunks c0..c0+3 and c0+4..c0+7), pre-scaled.
  v2f a0, a1;
  a0.x = qbase[row * Dh + c0 + 2 * hi]     * SCALE;
  a0.y = qbase[row * Dh + c0 + 2 * hi + 1] * SCALE;
  a1.x = qbase[row * Dh + c0 + 4 + 2 * hi] * SCALE;
  a1.y = qbase[row * Dh + c0 + 5 + 2 * hi] * SCALE;

  for (int kt = 0; kt < QT; ++kt) {
    const int k0 = kt * 16;
    const float* kbase = kmat + (size_t)k0 * Dh;
    const int n = row;  // B fragment column
    v2f b0, b1;
    b0.x = kbase[n * Dh + c0 + 2 * hi];
    b0.y = kbase[n * Dh + c0 + 2 * hi + 1];
    b1.x = kbase[n * Dh + c0 + 4 + 2 * hi];
    b1.y = kbase[n * Dh + c0 + 5 + 2 * hi];

    v8f acc = {};
    acc = WMMA4(a0, b0, acc);
    acc = WMMA4(a1, b1, acc);
#pragma unroll
    for (int r = 0; r < 8; ++r)
      S[(r + hi * 8) * L + k0 + n] = acc[r];   // already scaled
  }
  __syncthreads();

  // Row reductions: lane pair (row, row+16) splits the 512 columns.
  const float* srow = &S[row * L + hi * 256];
  float m = -1e30f;
#pragma unroll 8
  for (int k = 0; k < 256; ++k) m = fmaxf(m, srow[k]);
  m = fmaxf(m, __shfl_xor(m, 16));

  float Zp = 0.f, sq = 0.f, dse = 0.f;
#pragma unroll 8
  for (int k = 0; k < 256; ++k) {
    const float s = srow[k];
    const float e = __expf(s - m);
    Zp += e; sq += e * e; dse += e * s;
  }
  Zp  += __shfl_xor(Zp, 16);
  sq  += __shfl_xor(sq, 16);
  dse += __shfl_xor(dse, 16);

  float varr = 0.f, maxr = 0.f, entr = 0.f, hhir = 0.f;
  if (hi == 0) {
    const float invZ = 1.f / Zp;
    const float sumsq_p = sq * invZ * invZ;
    const float dot_ps  = dse * invZ;
    const size_t ridx = (((size_t)(b * H + h) * DEP + d) * L + q0 + row);
    rowM[ridx] = m;
    rowZ[ridx] = Zp;
    varr = (sumsq_p - (1.f / (float)L)) * (1.f / (float)(L - 1));
    maxr = invZ;                      // max p = exp(m-m)/Z
    entr = __logf(Zp) + m - dot_ps;   // -sum p log p
    hhir = sumsq_p;
  }
  // 32-lane tree reduction (hi lanes contribute 0)
#pragma unroll
  for (int off = 16; off >= 1; off >>= 1) {
    varr += __shfl_down(varr, off);
    maxr += __shfl_down(maxr, off);
    entr += __shfl_down(entr, off);
    hhir += __shfl_down(hhir, off);
  }
  if (lane == 0) {
    part[0 * NBLK1 + bid] = varr;
    part[1 * NBLK1 + bid] = maxr;
    part[2 * NBLK1 + bid] = entr;
    part[3 * NBLK1 + bid] = hhir;
  }
}

// ---------------------------------------------------------------------------
// Pass 2: reduce partials -> means -> min-max norm over D -> gate softmax.
// One wave: thread t handles (b = t/8, d = t%8).
// ---------------------------------------------------------------------------
__global__ void __launch_bounds__(32)
maw_pass2(const float* __restrict__ part, float* __restrict__ wts)
{
  const int t = threadIdx.x;
  const int b = t >> 3, d = t & 7;
  float s0 = 0.f, s1 = 0.f, s2 = 0.f, s3 = 0.f;
  for (int h = 0; h < H; ++h)
    for (int qt = 0; qt < QT; ++qt) {
      const int bid = ((b * H + h) * DEP + d) * QT + qt;
      s0 += part[0 * NBLK1 + bid];
      s1 += part[1 * NBLK1 + bid];
      s2 += part[2 * NBLK1 + bid];
      s3 += part[3 * NBLK1 + bid];
    }
  const float inv = 1.f / (float)(H * L);
  float v = s0 * inv, mx = s1 * inv, en = s2 * inv, hh = s3 * inv;

  auto nrm = [&](float x) -> float {
    float mn = x, mxv = x;
#pragma unroll
    for (int o = 1; o < 8; o <<= 1) {
      mn  = fminf(mn,  __shfl_xor(mn,  o, 8));
      mxv = fmaxf(mxv, __shfl_xor(mxv, o, 8));
    }
    return (x - mn) / (mxv - mn + 1e-8f);
  };
  const float score = 0.5f * nrm(v) + 0.3f * nrm(mx) + 0.2f * nrm(hh) - 0.4f * nrm(en);
  float a = 2.5f * score;  // alpha = 1 + 10*0.15
  float am = a;
#pragma unroll
  for (int o = 1; o < 8; o <<= 1) am = fmaxf(am, __shfl_xor(am, o, 8));
  const float e = __expf(a - am);
  float se = e;
#pragma unroll
  for (int o = 1; o < 8; o <<= 1) se += __shfl_xor(se, o, 8);
  wts[b * DEP + d] = e / se;
}

// ---------------------------------------------------------------------------
// Pass 3a: context = w_d * (p_d · V_d). Recompute p tiles, stage in LDS to
// re-fragment as WMMA A operand, multiply by V tile (8 valid columns).
// ---------------------------------------------------------------------------
__global__ void __launch_bounds__(32)
maw_pass3_ctx(const float* __restrict__ Q, const float* __restrict__ Kc,
              const float* __restrict__ V, const float* __restrict__ rowM,
              const float* __restrict__ rowZ, const float* __restrict__ wts,
              float* __restrict__ ctx)
{
  __shared__ float P[16 * 16];
  __shared__ float MM[16], ZZ[16];

  const unsigned bid = blockIdx.x;
  const int qt = bid & 31;
  const int d  = (bid >> 5) & 7;
  const int h  = (bid >> 8) & 15;
  const int b  = bid >> 12;
  const int q0 = qt * 16;
  const int c0 = d * DS;

  const int lane = threadIdx.x;
  const int row  = lane & 15;
  const int hi   = lane >> 4;
  const int n    = row;

  const size_t ridx = ((size_t)(b * H + h) * DEP + d) * L + q0;
  if (lane < 16) {
    MM[lane] = rowM[ridx + lane];
    ZZ[lane] = 1.f / rowZ[ridx + lane];
  }
  __syncthreads();

  float mreg[8], zreg[8];
#pragma unroll
  for (int r = 0; r < 8; ++r) { mreg[r] = MM[r + hi * 8]; zreg[r] = ZZ[r + hi * 8]; }

  const float* qbase = Q + ((size_t)(b * H + h) * L + q0) * Dh;
  v2f a0, a1;   // pre-scaled Q fragments
  a0.x = qbase[row * Dh + c0 + 2 * hi]     * SCALE;
  a0.y = qbase[row * Dh + c0 + 2 * hi + 1] * SCALE;
  a1.x = qbase[row * Dh + c0 + 4 + 2 * hi] * SCALE;
  a1.y = qbase[row * Dh + c0 + 5 + 2 * hi] * SCALE;

  const float* kmat = Kc + ((size_t)(b * H + h) * L) * Dh;
  const float* vmat = V  + ((size_t)(b * H + h) * L) * Dh;

  v8f cacc = {};
  for (int kt = 0; kt < QT; ++kt) {
    const int k0 = kt * 16;
    const float* kbase = kmat + (size_t)k0 * Dh;
    v2f b0, b1;
    b0.x = kbase[n * Dh + c0 + 2 * hi];
    b0.y = kbase[n * Dh + c0 + 2 * hi + 1];
    b1.x = kbase[n * Dh + c0 + 4 + 2 * hi];
    b1.y = kbase[n * Dh + c0 + 5 + 2 * hi];

    v8f sacc = {};
    sacc = WMMA4(a0, b0, sacc);
    sacc = WMMA4(a1, b1, sacc);
#pragma unroll
    for (int r = 0; r < 8; ++r)
      P[(r + hi * 8) * 16 + n] = __expf(sacc[r] - mreg[r]) * zreg[r];
    __syncthreads();

    const float* vbase = vmat + (size_t)k0 * Dh + c0;
#pragma unroll
    for (int j = 0; j < 4; ++j) {
      const int kk = j * 4;
      v2f pa, vb;
      pa.x = P[row * 16 + kk + 2 * hi];
      pa.y = P[row * 16 + kk + 2 * hi + 1];
      vb.x = (n < DS) ? vbase[(kk + 2 * hi) * Dh + n] : 0.f;
      vb.y = (n < DS) ? vbase[(kk + 2 * hi + 1) * Dh + n] : 0.f;
      cacc = WMMA4(pa, vb, cacc);
    }
    __syncthreads();
  }

  const float w = wts[b * DEP + d];
  if (n < DS) {
#pragma unroll
    for (int r = 0; r < 8; ++r)
      ctx[((size_t)(b * H + h) * L + q0 + r + hi * 8) * Dh + c0 + n] = w * cacc[r];
  }
}

// ---------------------------------------------------------------------------
// Pass 3b: attn_4d = sum_d w_d * p_d. One block per (b,h,q-tile); loops all
// 8 depths per k-tile, writes the mixed 16x16 tile once (no atomics).
// ---------------------------------------------------------------------------
__global__ void __launch_bounds__(32)
maw_pass3_attn(const float* __restrict__ Q, const float* __restrict__ Kc,
               const float* __restrict__ rowM, const float* __restrict__ rowZ,
               const float* __restrict__ wts, float* __restrict__ attn)
{
  __shared__ float MM[DEP * 16], ZZ[DEP * 16];

  const unsigned bid = blockIdx.x;
  const int qt = bid & 31;
  const int h  = (bid >> 5) & 15;
  const int b  = bid >> 9;
  const int q0 = qt * 16;

  const int lane = threadIdx.x;
  const int row  = lane & 15;
  const int hi   = lane >> 4;
  const int n    = row;

  for (int i = lane; i < DEP * 16; i += 32) {
    const int dd = i >> 4, rr = i & 15;
    const size_t ridx = ((size_t)(b * H + h) * DEP + dd) * L + q0 + rr;
    MM[i] = rowM[ridx];
    ZZ[i] = 1.f / rowZ[ridx];
  }
  __syncthreads();

  const float* qbase = Q + ((size_t)(b * H + h) * L + q0) * Dh;
  v2f aq0[DEP], aq1[DEP];   // pre-scaled Q fragments, all depths
  float wreg[DEP];
#pragma unroll
  for (int dd = 0; dd < DEP; ++dd) {
    const int c0 = dd * DS;
    aq0[dd].x = qbase[row * Dh + c0 + 2 * hi]     * SCALE;
    aq0[dd].y = qbase[row * Dh + c0 + 2 * hi + 1] * SCALE;
    aq1[dd].x = qbase[row * Dh + c0 + 4 + 2 * hi] * SCALE;
    aq1[dd].y = qbase[row * Dh + c0 + 5 + 2 * hi] * SCALE;
    wreg[dd]  = wts[b * DEP + dd];
  }

  const float* kmat = Kc + ((size_t)(b * H + h) * L) * Dh;
  for (int kt = 0; kt < QT; ++kt) {
    const int k0 = kt * 16;
    const float* kbase = kmat + (size_t)k0 * Dh;
    v8f facc = {};
#pragma unroll
    for (int dd = 0; dd < DEP; ++dd) {
      const int c0 = dd * DS;
      v2f b0, b1;
      b0.x = kbase[n * Dh + c0 + 2 * hi];
      b0.y = kbase[n * Dh + c0 + 2 * hi + 1];
      b1.x = kbase[n * Dh + c0 + 4 + 2 * hi];
      b1.y = kbase[n * Dh + c0 + 5 + 2 * hi];
      v8f sacc = {};
      sacc = WMMA4(aq0[dd], b0, sacc);
      sacc = WMMA4(aq1[dd], b1, sacc);
#pragma unroll
      for (int r = 0; r < 8; ++r) {
        const int rr = r + hi * 8;
        facc[r] += wreg[dd] * __expf(sacc[r] - MM[dd * 16 + rr]) * ZZ[dd * 16 + rr];
      }
    }
#pragma unroll
    for (int r = 0; r < 8; ++r)
      attn[((size_t)(b * H + h) * L + q0 + r + hi * 8) * L + k0 + n] = facc[r];
  }
}

// ---------------------------------------------------------------------------
extern "C" void kernel_launch(void* const* d_in, const int* in_sizes, int n_in,
                              void* d_out, int out_size, void* d_ws, size_t ws_size,
                              hipStream_t stream)
{
  const float* Q = (const float*)d_in[0];
  const float* K = (const float*)d_in[1];
  const float* V = (const float*)d_in[2];

  float* ctx_out  = (float*)d_out;                       // [B,H,L,Dh]
  float* attn_out = ctx_out + (size_t)B * H * L * Dh;    // [B,H,L,L]

  float* ws    = (float*)d_ws;
  float* rowM  = ws;                         // B*H*DEP*L = 262144
  float* rowZ  = rowM + (size_t)B * H * DEP * L;
  float* part  = rowZ + (size_t)B * H * DEP * L;         // 4 * NBLK1 = 65536
  float* wts   = part + (size_t)4 * NBLK1;               // B*DEP = 32

  maw_pass1<<<NBLK1, 32, 0, stream>>>(Q, K, rowM, rowZ, part);
  maw_pass2<<<1, 32, 0, stream>>>(part, wts);
  maw_pass3_ctx<<<NBLK1, 32, 0, stream>>>(Q, K, V, rowM, rowZ, wts, ctx_out);
  maw_pass3_attn<<<B * H * QT, 32, 0, stream>>>(Q, K, rowM, rowZ, wts, attn_out);
}